// FNO1d_37830071943755
// MI455X (gfx1250) — compile-verified
//
#include <hip/hip_runtime.h>
#include <math.h>

// ---------------------------------------------------------------------------
// FNO1d forward for MI455X (gfx1250), all matrix work on V_WMMA_F32_16X16X4_F32.
// B=64, C_IN=1, N=8192, WIDTH=64, MODES=16, NLAYERS=4.
//
// MODES << N  =>  replace rfft/irfft with dense GEMMs vs precomputed trig
// bases. h (134 MB) stays resident in the 192 MB L2; ~34 GFLOP total, f32
// matrix-pipe bound. Workspace requirement: ~141 MB (see offsets below).
//
// Global->LDS staging uses GLOBAL_LOAD_ASYNC_TO_LDS_B128 (ASYNCcnt-tracked,
// no VGPR round-trip) when the builtin is available, else scalar fallback.
// ---------------------------------------------------------------------------

#define NPTS   8192
#define BATCH  64
#define WIDTH  64
#define MODES  16
#define JJ     32            // 2*MODES (Re rows 0..15, Im rows 16..31)
#define NLAY   4
#define NSEG   8             // k_dft splits the n-reduction into 8 segments
#define SEGN   (NPTS / NSEG) // 1024
#define CHUNK  64            // LDS staging chunk along n

typedef float v2f __attribute__((ext_vector_type(2)));
typedef float v8f __attribute__((ext_vector_type(8)));

#if __has_builtin(__builtin_amdgcn_global_load_async_to_lds_b128)
#define HAVE_ASYNC_LDS 1
#else
#define HAVE_ASYNC_LDS 0
#endif

#if HAVE_ASYNC_LDS
// Builtin parameter type (from hipcc diagnostic): pointer to int4 vector.
typedef int ai4 __attribute__((vector_size(4 * sizeof(int))));
typedef __attribute__((address_space(1))) ai4* gp4_t;
typedef __attribute__((address_space(3))) ai4* lp4_t;
// One b128 transfer per lane: 16 bytes global -> LDS, tracked by ASYNCcnt.
__device__ __forceinline__ void async_ld_b128(const float* g, float* l) {
  __builtin_amdgcn_global_load_async_to_lds_b128((gp4_t)(g), (lp4_t)(l), 0, 0);
}
__device__ __forceinline__ void async_wait0() {
  asm volatile("s_wait_asynccnt 0x0" ::: "memory");
}
#endif

// Workspace layout (float offsets)
#define OFF_BFWD 0u                               // bfwdT [32][8192]  (1 MB)
#define OFF_BINV (OFF_BFWD + JJ * NPTS)           // binv  [32][8192]  (1 MB)
#define OFF_H    (OFF_BINV + JJ * NPTS)           // h [64][64][8192]  (134 MB)
#define OFF_XP   (OFF_H + (size_t)BATCH * WIDTH * NPTS)   // Xp [8][32][64][64] (4 MB)
#define OFF_Y    (OFF_XP + (size_t)NSEG * JJ * BATCH * WIDTH) // Yv [64][32][64] (0.5 MB)
// total = OFF_Y + 64*32*64 floats  ~= 141 MB

__device__ __forceinline__ v8f wmma_f32(v2f a, v2f b, v8f c) {
  // V_WMMA_F32_16X16X4_F32 : D = A(16x4) * B(4x16) + C(16x16), all f32.
  return __builtin_amdgcn_wmma_f32_16x16x4_f32(false, a, false, b, (short)0, c,
                                               false, false);
}

// ---------------------------------------------------------------------------
// Basis precompute.
//   bfwdT[j][n]: j<16 -> cos(2*pi*j*n/N); j>=16 -> -sin(2*pi*(j-16)*n/N)
//   binv [j][n]: j==0 -> 1/N; 1<=j<16 -> (2/N)cos; j==16 -> 0; j>16 -> -(2/N)sin
// ---------------------------------------------------------------------------
__global__ void k_basis(float* __restrict__ bfwdT, float* __restrict__ binv) {
  int idx = blockIdx.x * 256 + threadIdx.x;   // over JJ*NPTS
  int j = idx / NPTS, n = idx % NPTS;
  int k = j & 15;
  float th = (float)((k * n) & (NPTS - 1)) * (6.28318530717958647692f / (float)NPTS);
  float c = cosf(th), s = sinf(th);
  bfwdT[idx] = (j < 16) ? c : -s;
  float v;
  if (j == 0)        v = 1.0f / (float)NPTS;
  else if (j < 16)   v = (2.0f / (float)NPTS) * c;
  else if (j == 16)  v = 0.0f;                // Im(Y0) ignored by irfft
  else               v = -(2.0f / (float)NPTS) * s;
  binv[idx] = v;
}

// ---------------------------------------------------------------------------
// Lifting: h[b,w,n] = p_w[w,0]*x[b,0,n] + p_w[w,1]*(n/(N-1)) + p_b[w]
// ---------------------------------------------------------------------------
__global__ void k_lift(const float* __restrict__ x, const float* __restrict__ p_w,
                       const float* __restrict__ p_b, float* __restrict__ h) {
  size_t idx = (size_t)blockIdx.x * 256 + threadIdx.x;  // B*WIDTH*NPTS
  int n = (int)(idx & (NPTS - 1));
  size_t t = idx >> 13;
  int w = (int)(t & (WIDTH - 1));
  int b = (int)(t >> 6);
  float g = (float)n * (1.0f / 8191.0f);
  h[idx] = p_w[w * 2 + 0] * x[(size_t)b * NPTS + n] + p_w[w * 2 + 1] * g + p_b[w];
}

// ---------------------------------------------------------------------------
// Forward DFT (16 modes): partial X over one n-segment.
//   C[j,i] = sum_n bfwdT[j,n] * h[b,i,n],  one WG per (b, segment).
// 8 waves: (j-tile in {0,1}) x (i-tile in 0..3). K-loop chunks of 64 via LDS.
// ---------------------------------------------------------------------------
__global__ __launch_bounds__(256) void k_dft(const float* __restrict__ bfwdT,
                                             const float* __restrict__ h,
                                             float* __restrict__ Xp) {
  __shared__ __align__(16) float sA[JJ * CHUNK];     // basis chunk  [j][nn]  8 KB
  __shared__ __align__(16) float sB[WIDTH * CHUNK];  // h chunk      [i][nn] 16 KB
  int b = blockIdx.x / NSEG, seg = blockIdx.x % NSEG;
  int n0 = seg * SEGN;
  int tid = threadIdx.x, lane = tid & 31, wv = tid >> 5;
  int jt = wv & 1, it = wv >> 1;
  int mlo = lane & 15, khalf = (lane >> 4) * 2;
  v8f acc = {};
  for (int c = 0; c < SEGN / CHUNK; ++c) {
    int nc = n0 + c * CHUNK;
    __syncthreads();
#if HAVE_ASYNC_LDS
    // sA: 512 x b128 blocks (2/thread); sB: 1024 x b128 blocks (4/thread).
    for (int r = 0; r < 2; ++r) {
      int t2 = tid + r * 256;
      int j = t2 >> 4, q = (t2 & 15) * 4;
      async_ld_b128(&bfwdT[j * NPTS + nc + q], &sA[j * CHUNK + q]);
    }
    for (int r = 0; r < 4; ++r) {
      int t2 = tid + r * 256;
      int i = t2 >> 4, q = (t2 & 15) * 4;
      async_ld_b128(&h[((size_t)b * WIDTH + i) * NPTS + nc + q], &sB[i * CHUNK + q]);
    }
    async_wait0();
#else
    for (int t = tid; t < JJ * CHUNK; t += 256)
      sA[t] = bfwdT[(t >> 6) * NPTS + nc + (t & 63)];
    for (int t = tid; t < WIDTH * CHUNK; t += 256)
      sB[t] = h[((size_t)b * WIDTH + (t >> 6)) * NPTS + nc + (t & 63)];
#endif
    if (c + 1 < SEGN / CHUNK)  // gfx1250 global_prefetch_b8 for next chunk
      __builtin_prefetch(&h[((size_t)b * WIDTH + (tid >> 6)) * NPTS + nc + CHUNK], 0, 1);
    __syncthreads();
    for (int kk = 0; kk < CHUNK / 4; ++kk) {
      int kb = kk * 4 + khalf;
      v2f a, bb;
      a.x  = sA[(jt * 16 + mlo) * CHUNK + kb];
      a.y  = sA[(jt * 16 + mlo) * CHUNK + kb + 1];
      bb.x = sB[(it * 16 + mlo) * CHUNK + kb];
      bb.y = sB[(it * 16 + mlo) * CHUNK + kb + 1];
      acc = wmma_f32(a, bb, acc);
    }
  }
  for (int r = 0; r < 8; ++r) {   // C row map: VGPR r -> M=r (lanes<16) / r+8
    int j = jt * 16 + ((lane < 16) ? r : r + 8);
    int i = it * 16 + mlo;
    Xp[(((size_t)seg * JJ + j) * BATCH + b) * WIDTH + i] = acc[r];
  }
}

// ---------------------------------------------------------------------------
// Complex mode mixing, one WG per mode k (16 WGs):
//   Yr[b,o] = Xr*Wr - Xi*Wi ; Yi[b,o] = Xr*Wi + Xi*Wr   (64x64x64 GEMMs)
// Segment partials of X are reduced while filling LDS (deterministic order).
// f32 WMMA has no A-negate modifier -> negate Ai in VALU.
// ---------------------------------------------------------------------------
__global__ __launch_bounds__(256) void k_mix(const float* __restrict__ Xp,
                                             const float* __restrict__ wr,
                                             const float* __restrict__ wi,
                                             float* __restrict__ Yv, int layer) {
  __shared__ __align__(16) float sXr[BATCH * WIDTH];  // [b][i] 16 KB
  __shared__ __align__(16) float sXi[BATCH * WIDTH];
  __shared__ __align__(16) float sWr[WIDTH * WIDTH];  // [i][o] 16 KB
  __shared__ __align__(16) float sWi[WIDTH * WIDTH];
  int k = blockIdx.x, tid = threadIdx.x;
  for (int e = tid; e < BATCH * WIDTH; e += 256) {
    int b = e >> 6, i = e & 63;
    float sr = 0.f, si = 0.f;
    for (int s = 0; s < NSEG; ++s) {
      sr += Xp[(((size_t)s * JJ + k) * BATCH + b) * WIDTH + i];
      si += Xp[(((size_t)s * JJ + k + MODES) * BATCH + b) * WIDTH + i];
    }
    sXr[e] = sr; sXi[e] = si;
  }
  for (int e = tid; e < WIDTH * WIDTH; e += 256) {
    int i = e >> 6, o = e & 63;
    size_t off = (((size_t)layer * WIDTH + i) * WIDTH + o) * MODES + k;
    sWr[e] = wr[off];
    sWi[e] = wi[off];
  }
  __syncthreads();
  int lane = tid & 31, wv = tid >> 5;
  int mlo = lane & 15, khalf = (lane >> 4) * 2;
  for (int tt = 0; tt < 2; ++tt) {
    int tile = wv + tt * 8;
    int bt = tile >> 2, ot = tile & 3;
    v8f aR = {}, aI = {};
    for (int kk = 0; kk < WIDTH / 4; ++kk) {
      int kb = kk * 4 + khalf;
      int brow = bt * 16 + mlo, o = ot * 16 + mlo;
      v2f ar, ai, nai, br, bi;
      ar.x = sXr[brow * 64 + kb];  ar.y = sXr[brow * 64 + kb + 1];
      ai.x = sXi[brow * 64 + kb];  ai.y = sXi[brow * 64 + kb + 1];
      nai.x = -ai.x;               nai.y = -ai.y;
      br.x = sWr[kb * 64 + o];     br.y = sWr[(kb + 1) * 64 + o];
      bi.x = sWi[kb * 64 + o];     bi.y = sWi[(kb + 1) * 64 + o];
      aR = wmma_f32(ar, br, aR);
      aR = wmma_f32(nai, bi, aR);
      aI = wmma_f32(ar, bi, aI);
      aI = wmma_f32(ai, br, aI);
    }
    for (int r = 0; r < 8; ++r) {
      int bg = bt * 16 + ((lane < 16) ? r : r + 8);
      int o  = ot * 16 + mlo;
      Yv[((size_t)bg * JJ + k) * WIDTH + o]         = aR[r];
      Yv[((size_t)bg * JJ + k + MODES) * WIDTH + o] = aI[r];
    }
  }
}

// ---------------------------------------------------------------------------
// Fused inverse-DFT + pointwise conv + bias + exact GELU, in-place h update.
// One WG per (b, 64-wide n-tile): 8192 WGs. Single f32 accumulator chain:
//   acc = Y^T(64x32) * binv(32x64)  +  W(64x64) * h(64x64)
// ---------------------------------------------------------------------------
__global__ __launch_bounds__(256) void k_layer(const float* __restrict__ binv,
                                               const float* __restrict__ Yv,
                                               const float* __restrict__ w_w,
                                               const float* __restrict__ w_b,
                                               float* __restrict__ h, int layer) {
  __shared__ __align__(16) float sY [JJ * WIDTH];     // [j][o]   8 KB
  __shared__ __align__(16) float sBi[JJ * CHUNK];     // [j][nn]  8 KB
  __shared__ __align__(16) float sW [WIDTH * WIDTH];  // [o][i]  16 KB
  __shared__ __align__(16) float sH [WIDTH * CHUNK];  // [i][nn] 16 KB
  int b  = blockIdx.x >> 7;
  int n0 = (blockIdx.x & 127) * CHUNK;
  int tid = threadIdx.x;
#if HAVE_ASYNC_LDS
  for (int r = 0; r < 2; ++r) {          // sY: 512 b128 blocks (contiguous)
    int t2 = tid + r * 256;
    async_ld_b128(&Yv[(size_t)b * JJ * WIDTH + t2 * 4], &sY[t2 * 4]);
  }
  for (int r = 0; r < 2; ++r) {          // sBi: rows of binv
    int t2 = tid + r * 256;
    int j = t2 >> 4, q = (t2 & 15) * 4;
    async_ld_b128(&binv[j * NPTS + n0 + q], &sBi[j * CHUNK + q]);
  }
  for (int r = 0; r < 4; ++r) {          // sW: 1024 b128 blocks (contiguous)
    int t2 = tid + r * 256;
    async_ld_b128(&w_w[(size_t)layer * WIDTH * WIDTH + t2 * 4], &sW[t2 * 4]);
  }
  for (int r = 0; r < 4; ++r) {          // sH: rows of h
    int t2 = tid + r * 256;
    int i = t2 >> 4, q = (t2 & 15) * 4;
    async_ld_b128(&h[((size_t)b * WIDTH + i) * NPTS + n0 + q], &sH[i * CHUNK + q]);
  }
  async_wait0();
#else
  for (int t = tid; t < JJ * WIDTH; t += 256)    sY[t]  = Yv[(size_t)b * JJ * WIDTH + t];
  for (int t = tid; t < JJ * CHUNK; t += 256)    sBi[t] = binv[(t >> 6) * NPTS + n0 + (t & 63)];
  for (int t = tid; t < WIDTH * WIDTH; t += 256) sW[t]  = w_w[(size_t)layer * WIDTH * WIDTH + t];
  for (int t = tid; t < WIDTH * CHUNK; t += 256)
    sH[t] = h[((size_t)b * WIDTH + (t >> 6)) * NPTS + n0 + (t & 63)];
#endif
  __syncthreads();
  int lane = tid & 31, wv = tid >> 5;
  int mlo = lane & 15, khalf = (lane >> 4) * 2;
  for (int tt = 0; tt < 2; ++tt) {
    int tile = wv + tt * 8;
    int ot = tile >> 2, nt = tile & 3;
    int orow = ot * 16 + mlo, ncol = nt * 16 + mlo;
    v8f acc = {};
    for (int kk = 0; kk < JJ / 4; ++kk) {        // inverse DFT, K=32 (j)
      int kb = kk * 4 + khalf;
      v2f a, bb;
      a.x  = sY[kb * 64 + orow];       a.y  = sY[(kb + 1) * 64 + orow];
      bb.x = sBi[kb * 64 + ncol];      bb.y = sBi[(kb + 1) * 64 + ncol];
      acc = wmma_f32(a, bb, acc);
    }
    for (int kk = 0; kk < WIDTH / 4; ++kk) {     // pointwise conv, K=64 (i)
      int kb = kk * 4 + khalf;
      v2f a, bb;
      a.x  = sW[orow * 64 + kb];       a.y  = sW[orow * 64 + kb + 1];
      bb.x = sH[kb * 64 + ncol];       bb.y = sH[(kb + 1) * 64 + ncol];
      acc = wmma_f32(a, bb, acc);
    }
    for (int r = 0; r < 8; ++r) {
      int og = ot * 16 + ((lane < 16) ? r : r + 8);
      int ng = n0 + nt * 16 + mlo;
      float v = acc[r] + w_b[layer * WIDTH + og];
      v = 0.5f * v * (1.0f + erff(v * 0.70710678118654752f));  // exact GELU
      h[((size_t)b * WIDTH + og) * NPTS + ng] = v;
    }
  }
}

// ---------------------------------------------------------------------------
// Projection: out[b,0,n] = q_b + sum_w q_w[w] * h[b,w,n]
// ---------------------------------------------------------------------------
__global__ void k_proj(const float* __restrict__ h, const float* __restrict__ q_w,
                       const float* __restrict__ q_b, float* __restrict__ out) {
  size_t idx = (size_t)blockIdx.x * 256 + threadIdx.x;  // B*NPTS
  int n = (int)(idx & (NPTS - 1));
  int b = (int)(idx >> 13);
  float s = q_b[0];
#pragma unroll 8
  for (int w = 0; w < WIDTH; ++w)
    s += q_w[w] * h[((size_t)b * WIDTH + w) * NPTS + n];
  out[idx] = s;
}

// ---------------------------------------------------------------------------
extern "C" void kernel_launch(void* const* d_in, const int* in_sizes, int n_in,
                              void* d_out, int out_size, void* d_ws, size_t ws_size,
                              hipStream_t stream) {
  (void)in_sizes; (void)n_in; (void)out_size; (void)ws_size;
  const float* x    = (const float*)d_in[0];
  const float* p_w  = (const float*)d_in[1];
  const float* p_b  = (const float*)d_in[2];
  const float* scwr = (const float*)d_in[3];
  const float* scwi = (const float*)d_in[4];
  const float* w_w  = (const float*)d_in[5];
  const float* w_b  = (const float*)d_in[6];
  const float* q_w  = (const float*)d_in[7];
  const float* q_b  = (const float*)d_in[8];

  float* ws    = (float*)d_ws;
  float* bfwdT = ws + OFF_BFWD;
  float* binv  = ws + OFF_BINV;
  float* h     = ws + OFF_H;
  float* Xp    = ws + OFF_XP;
  float* Yv    = ws + OFF_Y;

  k_basis<<<(JJ * NPTS) / 256, 256, 0, stream>>>(bfwdT, binv);
  k_lift<<<(BATCH * WIDTH * NPTS) / 256, 256, 0, stream>>>(x, p_w, p_b, h);
  for (int l = 0; l < NLAY; ++l) {
    k_dft<<<BATCH * NSEG, 256, 0, stream>>>(bfwdT, h, Xp);
    k_mix<<<MODES, 256, 0, stream>>>(Xp, scwr, scwi, Yv, l);
    k_layer<<<BATCH * (NPTS / CHUNK), 256, 0, stream>>>(binv, Yv, w_w, w_b, h, l);
  }
  k_proj<<<(BATCH * NPTS) / 256, 256, 0, stream>>>(h, q_w, q_b, (float*)d_out);
}